// GHM_Loss_22162031247441
// MI455X (gfx1250) — compile-verified
//
#include <hip/hip_runtime.h>
#include <math.h>

#define BINS     10
#define NBLOCKS  1024
#define NTHREADS 256

typedef float v2f __attribute__((ext_vector_type(2)));
typedef float v8f __attribute__((ext_vector_type(8)));

// Full-precision fp32 wave32 reduction using V_WMMA_F32_16X16X4_F32.
// A: lane L(0..15) -> A[L][0]=p, A[L][1]=0 ; lane L(16..31) -> A[L-16][2]=p, A[L-16][3]=0
// B: all ones (4x16) -> D[m][n] = p_m + p_{m+16} for every n.
// Lane n<16 holds D[0..7][n] in its 8 C VGPRs, lane n>=16 holds D[8..15][n-16];
// summing the 8 components then xor-shuffling across the half-waves gives the wave total.
__device__ __forceinline__ float wave_reduce_wmma(float p) {
    v2f a; a[0] = p;    a[1] = 0.0f;
    v2f b; b[0] = 1.0f; b[1] = 1.0f;
    v8f c = {};
    v8f d = __builtin_amdgcn_wmma_f32_16x16x4_f32(
        /*neg_a=*/false, a, /*neg_b=*/false, b,
        /*c_mod=*/(short)0, c, /*reuse_a=*/false, /*reuse_b=*/false);
    float t = 0.0f;
#pragma unroll
    for (int j = 0; j < 8; ++j) t += d[j];
    t += __shfl_xor(t, 16, 32);   // combine M=0..7 half with M=8..15 half
    return t;                     // every lane now holds the wave32 sum
}

__device__ __forceinline__ int bin_of(float xc, float tc) {
    float sig = 1.0f / (1.0f + expf(-xc));
    float g   = fabsf(sig - tc);
    int   b   = (int)(g * 9.9999f);     // floor(g * (BINS - 1e-4))
    b = b < 0 ? 0 : (b > BINS - 1 ? BINS - 1 : b);
    return b;
}

// ---- kernel 0: zero global bins (graph-capture safe init) ----
__global__ void ghm_init(unsigned int* __restrict__ gbins) {
    if (threadIdx.x < BINS) gbins[threadIdx.x] = 0u;
}

// ---- kernel 1: global 10-bin histogram of gradient norms ----
__global__ void ghm_hist(const float4* __restrict__ x, const int* __restrict__ tgt,
                         unsigned int* __restrict__ gbins, int nrows) {
    __shared__ unsigned int hist[NTHREADS][BINS];   // private per-thread: no LDS atomics
    const int tid = threadIdx.x;
#pragma unroll
    for (int b = 0; b < BINS; ++b) hist[tid][b] = 0u;

    const int stride = gridDim.x * blockDim.x;
    for (int i = blockIdx.x * blockDim.x + tid; i < nrows; i += stride) {
        const float4 xv = x[i];
        const int    t  = tgt[i];
        const float  xs[4] = {xv.x, xv.y, xv.z, xv.w};
#pragma unroll
        for (int cc = 0; cc < 4; ++cc) {
            const float tc = (t == cc) ? 1.0f : 0.0f;
            hist[tid][bin_of(xs[cc], tc)]++;
        }
    }
    __syncthreads();
    // tree-reduce the 256 private histograms
    for (int off = NTHREADS / 2; off > 0; off >>= 1) {
        if (tid < off) {
#pragma unroll
            for (int b = 0; b < BINS; ++b) hist[tid][b] += hist[tid + off][b];
        }
        __syncthreads();
    }
    if (tid < BINS) atomicAdd(&gbins[tid], hist[0][tid]);  // integer: deterministic
}

// ---- kernel 2: per-bin weights beta[b] = Ntot / clip(count*nonempty, 1e-4) ----
__global__ void ghm_beta(const unsigned int* __restrict__ gbins,
                         float* __restrict__ beta, float ntotal) {
    if (threadIdx.x == 0) {
        int nonempty = 0;
#pragma unroll
        for (int b = 0; b < BINS; ++b) nonempty += (gbins[b] > 0u) ? 1 : 0;
        const float ne = (float)nonempty;
#pragma unroll
        for (int b = 0; b < BINS; ++b) {
            float gd = fmaxf((float)gbins[b] * ne, 0.0001f);
            beta[b] = ntotal / gd;
        }
    }
}

// ---- kernel 3: weighted BCE partial sums (WMMA wave reduction) ----
__global__ void ghm_loss(const float4* __restrict__ x, const int* __restrict__ tgt,
                         const float* __restrict__ beta, float* __restrict__ partials,
                         int nrows) {
    __shared__ float sbeta[BINS];
    __shared__ float wsum[NTHREADS / 32];
    const int tid = threadIdx.x;
    if (tid < BINS) sbeta[tid] = beta[tid];
    __syncthreads();

    float acc = 0.0f;
    const int stride = gridDim.x * blockDim.x;
    for (int i = blockIdx.x * blockDim.x + tid; i < nrows; i += stride) {
        const float4 xv = x[i];
        const int    t  = tgt[i];
        const float  xs[4] = {xv.x, xv.y, xv.z, xv.w};
#pragma unroll
        for (int cc = 0; cc < 4; ++cc) {
            const float xc = xs[cc];
            const float tc = (t == cc) ? 1.0f : 0.0f;
            const float w  = sbeta[bin_of(xc, tc)];
            const float bce = fmaxf(xc, 0.0f) - xc * tc + log1pf(expf(-fabsf(xc)));
            acc += w * bce;
        }
    }

    // wave32 reduction on the matrix pipe (EXEC is all-ones here: full block, reconverged)
    const float wtot = wave_reduce_wmma(acc);
    const int lane = tid & 31, wave = tid >> 5;
    if (lane == 0) wsum[wave] = wtot;
    __syncthreads();
    if (tid == 0) {
        float s = 0.0f;
#pragma unroll
        for (int w = 0; w < NTHREADS / 32; ++w) s += wsum[w];
        partials[blockIdx.x] = s;
    }
}

// ---- kernel 4: deterministic fixed-order final sum + mean ----
__global__ void ghm_final(const float* __restrict__ partials, float* __restrict__ out,
                          float inv_ntotal) {
    __shared__ float s[NTHREADS];
    const int tid = threadIdx.x;
    float a = 0.0f;
    for (int i = tid; i < NBLOCKS; i += NTHREADS) a += partials[i];
    s[tid] = a;
    __syncthreads();
    for (int off = NTHREADS / 2; off > 0; off >>= 1) {
        if (tid < off) s[tid] += s[tid + off];
        __syncthreads();
    }
    if (tid == 0) out[0] = s[0] * inv_ntotal;
}

extern "C" void kernel_launch(void* const* d_in, const int* in_sizes, int n_in,
                              void* d_out, int out_size, void* d_ws, size_t ws_size,
                              hipStream_t stream) {
    const float4* x   = (const float4*)d_in[0];  // [N,4] fp32, row = float4
    const int*    tgt = (const int*)d_in[1];     // [N] int
    const int nrows   = in_sizes[1];
    const float ntotal = (float)nrows * 4.0f;    // 16e6 < 2^24: exact in fp32

    unsigned int* gbins    = (unsigned int*)d_ws;                 // 10 x u32
    float*        beta     = (float*)((char*)d_ws + 64);          // 10 x f32
    float*        partials = (float*)((char*)d_ws + 128);         // NBLOCKS x f32
    float*        out      = (float*)d_out;

    ghm_init <<<1, 32, 0, stream>>>(gbins);
    ghm_hist <<<NBLOCKS, NTHREADS, 0, stream>>>(x, tgt, gbins, nrows);
    ghm_beta <<<1, 32, 0, stream>>>(gbins, beta, ntotal);
    ghm_loss <<<NBLOCKS, NTHREADS, 0, stream>>>(x, tgt, beta, partials, nrows);
    ghm_final<<<1, NTHREADS, 0, stream>>>(partials, out, 1.0f / ntotal);
}